// Conv1dBlock_22402549416651
// MI455X (gfx1250) — compile-verified
//
#include <hip/hip_runtime.h>
#include <math.h>

// Problem constants (match reference)
#define BB 512
#define CC 256
#define TT 256
#define EE 8
#define KK 5
#define GG 8
#define CPG 32                 // channels per group = CC/GG
#define KJ (CC*KK)             // flattened reduction length = 1280
#define CI_TILE 64             // Cin rows staged in LDS per stage
#define NSTAGE (CC/CI_TILE)    // 4
#define JSPAN (CI_TILE*KK)     // 320 j-values per stage
#define KSTEPS (JSPAN/4)       // 80 WMMA K-steps per stage
#define NPER (KSTEPS/5)        // 16 periods of 5 steps
#define TCHUNK 64              // T columns per chunk
#define NCHUNK (TT/TCHUNK)     // 4
#define XS_W (TCHUNK+KK-1)     // 68 staged columns (halo 2 each side)
#define YS_W (TT+4)            // 260: padded row stride kills LDS bank conflicts
#define PSTRIDE (4*XS_W)       // 272: LDS float advance per 5-step period

typedef float v2f __attribute__((ext_vector_type(2)));
typedef float v8f __attribute__((ext_vector_type(8)));

__global__ __launch_bounds__(256) void conv_gn_mish_kernel(
    const float* __restrict__ x,       // [B, C, T]
    const int*   __restrict__ expert,  // [B]
    const float* __restrict__ W,       // [E, C, C, K]  -> [E, C, 1280] flat
    const float* __restrict__ bias,    // [E, C]
    const float* __restrict__ gamma,   // [E, C]
    const float* __restrict__ beta,    // [E, C]
    float* __restrict__ out)           // [B, C, T]
{
    __shared__ float xs[CI_TILE * XS_W];   // x stage (reused as reduce scratch)
    __shared__ float ys[CPG * YS_W];       // conv+bias result for this group
    __shared__ float bias_s[CPG], g_s[CPG], b_s[CPG];

    const int tid  = threadIdx.x;
    const int bidx = blockIdx.x;           // sample
    const int g    = blockIdx.y;           // group
    const int co0  = g * CPG;
    const int e    = expert[bidx];

    const int lane = tid & 31;
    const int l15  = lane & 15;
    const int hl   = lane >> 4;            // half-wave: 0 or 1
    const int wid  = tid >> 5;             // wave id 0..7
    const int m    = wid & 1;              // which 16-row half of the 32 channels
    const int n    = wid >> 1;             // which 16-col chunk of the 64 T cols

    if (tid < CPG) {
        bias_s[tid] = bias [e * CC + co0 + tid];
        g_s[tid]    = gamma[e * CC + co0 + tid];
        b_s[tid]    = beta [e * CC + co0 + tid];
    }
    __syncthreads();

    // A-matrix (W) row pointer for this lane: row = co0 + m*16 + l15,
    // lanes 16-31 carry K = j0+2, j0+3  (32-bit A layout, 16x4 tile).
    const float* Wrow = W + ((size_t)e * CC + (size_t)(co0 + m * 16 + l15)) * KJ
                          + 2 * hl;
    const float* xrow = x + (size_t)bidx * CC * TT;

    // Per-lane constant within-period cumulative LDS offsets (depend on hl only).
    // tracker0 starts k=2*hl, tracker1 starts k=2*hl+1; j advances by 4/step,
    // period = 5 steps (ci += 4, k returns), period stride = 4*XS_W = 272.
    const int c0[5] = { 0, hl ? 67 : 4,  hl ? 134 : 71, 138,            205 };
    const int c1[5] = { 0, 67,           hl ? 134 : 71, hl ? 201 : 138, 205 };
    const int a0i = n * 16 + l15 + 2 * hl;       // ci=0, col base + k_init
    const int a1i = a0i + 1;

    #pragma unroll 1
    for (int tc = 0; tc < NCHUNK; ++tc) {
        // Fold conv bias into one chain; second chain starts at zero.
        v8f acc0, acc1;
        #pragma unroll
        for (int r = 0; r < 8; ++r) { acc0[r] = bias_s[m * 16 + 8 * hl + r]; acc1[r] = 0.0f; }

        #pragma unroll 1
        for (int stage = 0; stage < NSTAGE; ++stage) {
            __syncthreads();               // xs reuse guard
            // Stage x[ci0..ci0+64) x [tc*64-2, tc*64+66) with zero padding.
            for (int i = tid; i < CI_TILE * XS_W; i += 256) {
                int ci = i / XS_W;
                int c  = i - ci * XS_W;
                int t  = tc * TCHUNK - (KK / 2) + c;
                float v = 0.0f;
                if (t >= 0 && t < TT)
                    v = xrow[(size_t)(stage * CI_TILE + ci) * TT + t];
                xs[i] = v;
            }
            __syncthreads();

            int p0[5], p1[5];
            #pragma unroll
            for (int i = 0; i < 5; ++i) { p0[i] = a0i + c0[i]; p1[i] = a1i + c1[i]; }

            const float* wp = Wrow + stage * JSPAN;

            // ---- software pipeline: double-buffered A (global) + B (LDS) ----
            v2f a_buf[2][5];
            v2f b_buf[2][5];
            #pragma unroll
            for (int i = 0; i < 5; ++i) {        // preload period 0
                a_buf[0][i] = *(const v2f*)(wp + 4 * i);
                b_buf[0][i][0] = xs[p0[i]];
                b_buf[0][i][1] = xs[p1[i]];
                p0[i] += PSTRIDE; p1[i] += PSTRIDE;
            }
            wp += 20;

            #pragma unroll 2
            for (int per = 0; per < NPER - 1; ++per) {
                const int cur = per & 1, nxt = cur ^ 1;
                #pragma unroll
                for (int i = 0; i < 5; ++i) {    // prefetch period per+1
                    a_buf[nxt][i] = *(const v2f*)(wp + 4 * i);
                    b_buf[nxt][i][0] = xs[p0[i]];
                    b_buf[nxt][i][1] = xs[p1[i]];
                    p0[i] += PSTRIDE; p1[i] += PSTRIDE;
                }
                wp += 20;
                #pragma unroll
                for (int i = 0; i < 5; ++i) {    // compute period per
                    if (i & 1)
                        acc1 = __builtin_amdgcn_wmma_f32_16x16x4_f32(
                                 false, a_buf[cur][i], false, b_buf[cur][i],
                                 (short)0, acc1, false, false);
                    else
                        acc0 = __builtin_amdgcn_wmma_f32_16x16x4_f32(
                                 false, a_buf[cur][i], false, b_buf[cur][i],
                                 (short)0, acc0, false, false);
                }
            }
            {   // peeled last period (no prefetch -> no out-of-bounds reads)
                const int cur = (NPER - 1) & 1;
                #pragma unroll
                for (int i = 0; i < 5; ++i) {
                    if (i & 1)
                        acc1 = __builtin_amdgcn_wmma_f32_16x16x4_f32(
                                 false, a_buf[cur][i], false, b_buf[cur][i],
                                 (short)0, acc1, false, false);
                    else
                        acc0 = __builtin_amdgcn_wmma_f32_16x16x4_f32(
                                 false, a_buf[cur][i], false, b_buf[cur][i],
                                 (short)0, acc0, false, false);
                }
            }
        }

        // Spill the 16x16 tile into ys (C/D layout: row = r + 8*hl, col = l15).
        #pragma unroll
        for (int r = 0; r < 8; ++r)
            ys[(m * 16 + 8 * hl + r) * YS_W + tc * TCHUNK + n * 16 + l15] =
                acc0[r] + acc1[r];
    }
    __syncthreads();

    // ---- GroupNorm statistics over 32 x 256 elements (deterministic tree) ----
    float s = 0.0f, sq = 0.0f;
    for (int i = tid; i < CPG * TT; i += 256) {
        int row = i >> 8;                  // /256
        int col = i & 255;
        float v = ys[row * YS_W + col];
        s  += v;
        sq += v * v;
    }
    xs[tid]       = s;                     // reuse xs as scratch
    xs[256 + tid] = sq;
    __syncthreads();
    for (int off = 128; off > 0; off >>= 1) {
        if (tid < off) {
            xs[tid]       += xs[tid + off];
            xs[256 + tid] += xs[256 + tid + off];
        }
        __syncthreads();
    }
    const float inv_n = 1.0f / (float)(CPG * TT);
    float mean = xs[0] * inv_n;
    float var  = xs[256] * inv_n - mean * mean;
    float rstd = rsqrtf(var + 1e-5f);

    // ---- Normalize + affine + Mish, write out ----
    for (int i = tid; i < CPG * TT; i += 256) {
        int row = i >> 8;
        int col = i & 255;
        float v = (ys[row * YS_W + col] - mean) * rstd;
        v = v * g_s[row] + b_s[row];
        float sp = log1pf(expf(v));        // exp->inf => tanh->1 => result v (safe)
        float r  = v * tanhf(sp);
        out[((size_t)bidx * CC + (size_t)(co0 + row)) * TT + col] = r;
    }
}

extern "C" void kernel_launch(void* const* d_in, const int* in_sizes, int n_in,
                              void* d_out, int out_size, void* d_ws, size_t ws_size,
                              hipStream_t stream) {
    (void)in_sizes; (void)n_in; (void)d_ws; (void)ws_size; (void)out_size;
    const float* x  = (const float*)d_in[0];
    const int*   ei = (const int*)  d_in[1];
    const float* W  = (const float*)d_in[2];
    const float* b  = (const float*)d_in[3];
    const float* gm = (const float*)d_in[4];
    const float* bt = (const float*)d_in[5];
    float* out = (float*)d_out;

    dim3 grid(BB, GG);                     // one block per (sample, group)
    conv_gn_mish_kernel<<<grid, 256, 0, stream>>>(x, ei, W, b, gm, bt, out);
}